// GAT_body_87265145520541
// MI455X (gfx1250) — compile-verified
//
#include <hip/hip_runtime.h>

// ---------------- problem constants (match reference) ----------------
#define N_NODES 50000
#define N_EDGES 800000
#define NFEAT   128
#define HEADS   8
#define NHID    16
#define H1DIM   (HEADS * NHID)   // 128
#define SLOPE   0.2f

typedef __attribute__((ext_vector_type(2))) float v2f;
typedef __attribute__((ext_vector_type(8))) float v8f;

// ---------------- helpers ----------------
__device__ __forceinline__ float lrelu(float x) { return x > 0.0f ? x : SLOPE * x; }

// order-preserving float <-> uint encoding so atomicMax(u32) == float max
__device__ __forceinline__ unsigned f2ord(float f) {
    unsigned u = __float_as_uint(f);
    return (u & 0x80000000u) ? ~u : (u | 0x80000000u);
}
__device__ __forceinline__ float ord2f(unsigned e) {
    unsigned u = (e & 0x80000000u) ? (e ^ 0x80000000u) : ~e;
    return __uint_as_float(u);
}

// ---------------- Kernel 1: fp32 WMMA GEMM  C[nRows,nCols] = A[nRows,128] * B[128,nCols]
// One 16x16 output tile per wave32, K stepped by 4 via V_WMMA_F32_16X16X4_F32.
// A frag (16x4 f32): lane(l16)=M, VGPR v holds K = v + 2*(lane>=16)
// B frag (4x16 f32): lane(l16)=N, VGPR v holds K = v + 2*(lane>=16)
// C/D (16x16 f32):   VGPR r = row r + 8*(lane>=16), lane(l16)=N
__global__ void gat_gemm_wmma_f32(const float* __restrict__ A,
                                  const float* __restrict__ B,
                                  float* __restrict__ C,
                                  int nRows, int nCols) {
    const int lane = threadIdx.x & 31;
    const int wave = threadIdx.x >> 5;
    const int nTilesN = nCols >> 4;
    const int nTilesM = nRows >> 4;          // nRows is a multiple of 16 here (50000)
    const int tile = blockIdx.x * (blockDim.x >> 5) + wave;
    if (tile >= nTilesM * nTilesN) return;   // wave-uniform exit: EXEC stays all-ones for WMMA

    const int m0   = (tile / nTilesN) << 4;
    const int n0   = (tile % nTilesN) << 4;
    const int half = lane >> 4;              // 0: lanes 0-15, 1: lanes 16-31
    const int l16  = lane & 15;

    const float* __restrict__ arow = A + (size_t)(m0 + l16) * NFEAT;

    v8f acc = {};
    for (int k0 = 0; k0 < NFEAT; k0 += 4) {
        const int ka = k0 + 2 * half;        // even -> 8B aligned
        const float2 av = *reinterpret_cast<const float2*>(arow + ka);
        v2f a; a.x = av.x; a.y = av.y;
        v2f b;
        b.x = B[(size_t)ka       * nCols + n0 + l16];
        b.y = B[(size_t)(ka + 1) * nCols + n0 + l16];
        // D = A*B + C  (8 args: neg_a, A, neg_b, B, c_mod, C, reuse_a, reuse_b)
        acc = __builtin_amdgcn_wmma_f32_16x16x4_f32(false, a, false, b,
                                                    (short)0, acc, false, false);
    }
#pragma unroll
    for (int r = 0; r < 8; ++r) {
        C[(size_t)(m0 + r + 8 * half) * nCols + n0 + l16] = acc[r];
    }
}

// ---------------- Kernel 2: per-(node,head) attention logits + emax seeded with self-loop
template <int H>
__global__ void gat_attn_coef(const float* __restrict__ h,       // [N, H*16]
                              const float* __restrict__ a_src,   // [H,16]
                              const float* __restrict__ a_dst,   // [H,16]
                              float* __restrict__ al_s,          // [N*H]
                              float* __restrict__ al_d,          // [N*H]
                              unsigned* __restrict__ emax_bits)  // [N*H]
{
    const int idx = blockIdx.x * blockDim.x + threadIdx.x;       // n*H + hd
    if (idx >= N_NODES * H) return;
    const int hd = idx % H;
    const float4* __restrict__ hp  = reinterpret_cast<const float4*>(h + (size_t)idx * NHID);
    const float4* __restrict__ asp = reinterpret_cast<const float4*>(a_src + hd * NHID);
    const float4* __restrict__ adp = reinterpret_cast<const float4*>(a_dst + hd * NHID);
    float s = 0.0f, d = 0.0f;
#pragma unroll
    for (int i = 0; i < 4; ++i) {
        const float4 hv = hp[i], av = asp[i], dv = adp[i];
        s += hv.x * av.x + hv.y * av.y + hv.z * av.z + hv.w * av.w;
        d += hv.x * dv.x + hv.y * dv.y + hv.z * dv.z + hv.w * dv.w;
    }
    al_s[idx] = s;
    al_d[idx] = d;
    emax_bits[idx] = f2ord(lrelu(s + d));    // self-loop edge seeds the segment max
}

// ---------------- Kernel 3: segment max over incoming edges (atomicMax on ordered bits)
template <int H>
__global__ void gat_edge_max(const int* __restrict__ src, const int* __restrict__ dst,
                             const float* __restrict__ al_s, const float* __restrict__ al_d,
                             unsigned* __restrict__ emax_bits)
{
    const int t = blockIdx.x * blockDim.x + threadIdx.x;         // e*H + hd
    if (t >= N_EDGES * H) return;
    const int e = t / H, hd = t % H;
    const int s = src[e], d = dst[e];
    const float val = lrelu(al_s[s * H + hd] + al_d[d * H + hd]);
    atomicMax(&emax_bits[d * H + hd], f2ord(val));
}

// ---------------- Kernel 4: init denom/num with the self-loop term (re-init every call)
template <int H>
__global__ void gat_init_sums(const float* __restrict__ h,
                              const float* __restrict__ al_s, const float* __restrict__ al_d,
                              const unsigned* __restrict__ emax_bits,
                              float* __restrict__ denom,         // [N*H]
                              float* __restrict__ num)           // [N, H*16]
{
    const int idx = blockIdx.x * blockDim.x + threadIdx.x;       // n*H + hd
    if (idx >= N_NODES * H) return;
    const float emax = ord2f(emax_bits[idx]);
    const float ex = __expf(lrelu(al_s[idx] + al_d[idx]) - emax);
    denom[idx] = ex;
    const float4* __restrict__ hp = reinterpret_cast<const float4*>(h + (size_t)idx * NHID);
    float4* __restrict__ np = reinterpret_cast<float4*>(num + (size_t)idx * NHID);
#pragma unroll
    for (int i = 0; i < 4; ++i) {
        const float4 v = hp[i];
        np[i] = make_float4(v.x * ex, v.y * ex, v.z * ex, v.w * ex);
    }
}

// ---------------- Kernel 5: edge aggregation (the bandwidth-bound hot loop)
template <int H>
__global__ void gat_edge_agg(const int* __restrict__ src, const int* __restrict__ dst,
                             const float* __restrict__ h,
                             const float* __restrict__ al_s, const float* __restrict__ al_d,
                             const unsigned* __restrict__ emax_bits,
                             float* __restrict__ denom, float* __restrict__ num)
{
    const int t = blockIdx.x * blockDim.x + threadIdx.x;         // e*H + hd
    if (t >= N_EDGES * H) return;
    const int e = t / H, hd = t % H;
    const int s = src[e], d = dst[e];
    const int si = s * H + hd, di = d * H + hd;
    const float val = lrelu(al_s[si] + al_d[di]);
    const float ex = __expf(val - ord2f(emax_bits[di]));
    atomicAdd(&denom[di], ex);
    const float4* __restrict__ hp = reinterpret_cast<const float4*>(h + (size_t)si * NHID);
    float* __restrict__ np = num + (size_t)di * NHID;
#pragma unroll
    for (int i = 0; i < 4; ++i) {
        const float4 v = hp[i];
        atomicAdd(np + 4 * i + 0, ex * v.x);
        atomicAdd(np + 4 * i + 1, ex * v.y);
        atomicAdd(np + 4 * i + 2, ex * v.z);
        atomicAdd(np + 4 * i + 3, ex * v.w);
    }
}

// ---------------- Kernel 6: out = relu(num/denom + bias)
template <int H>
__global__ void gat_finalize(const float* __restrict__ num, const float* __restrict__ denom,
                             const float* __restrict__ bias, float* __restrict__ out)
{
    const int idx = blockIdx.x * blockDim.x + threadIdx.x;       // over N * H*16
    if (idx >= N_NODES * H * NHID) return;
    const int col = idx % (H * NHID);
    const int nh  = idx >> 4;                                    // (n*H + hd)
    const float v = num[idx] / denom[nh] + bias[col];
    out[idx] = v > 0.0f ? v : 0.0f;
}

// ---------------- host side ----------------
static inline int cdiv(long long a, int b) { return (int)((a + b - 1) / b); }

extern "C" void kernel_launch(void* const* d_in, const int* in_sizes, int n_in,
                              void* d_out, int out_size, void* d_ws, size_t ws_size,
                              hipStream_t stream) {
    (void)in_sizes; (void)n_in; (void)out_size; (void)ws_size;

    const float* x   = (const float*)d_in[0];
    const int*   ei  = (const int*)  d_in[1];   // [2,E]: src then dst
    const float* W1  = (const float*)d_in[2];
    const float* as1 = (const float*)d_in[3];
    const float* ad1 = (const float*)d_in[4];
    const float* b1  = (const float*)d_in[5];
    const float* W2  = (const float*)d_in[6];
    const float* as2 = (const float*)d_in[7];
    const float* ad2 = (const float*)d_in[8];
    const float* b2  = (const float*)d_in[9];
    const int* src  = ei;
    const int* dstp = ei + N_EDGES;
    float* out = (float*)d_out;

    // workspace carve-up (floats), ~65 MB total, all re-initialized each call
    float* ws = (float*)d_ws;
    size_t o = 0;
    float*    h1     = ws + o; o += (size_t)N_NODES * H1DIM;   // 6.4M
    float*    num1   = ws + o; o += (size_t)N_NODES * H1DIM;   // 6.4M (becomes layer-2 input)
    float*    als1   = ws + o; o += (size_t)N_NODES * HEADS;
    float*    ald1   = ws + o; o += (size_t)N_NODES * HEADS;
    unsigned* emax1  = (unsigned*)(ws + o); o += (size_t)N_NODES * HEADS;
    float*    denom1 = ws + o; o += (size_t)N_NODES * HEADS;
    float*    h2     = ws + o; o += (size_t)N_NODES * NHID;
    float*    num2   = ws + o; o += (size_t)N_NODES * NHID;
    float*    als2   = ws + o; o += (size_t)N_NODES;
    float*    ald2   = ws + o; o += (size_t)N_NODES;
    unsigned* emax2  = (unsigned*)(ws + o); o += (size_t)N_NODES;
    float*    denom2 = ws + o; o += (size_t)N_NODES;

    const int B = 256;  // 8 wave32s per block

    // ---------------- layer 1: GATConv(128 -> 16, heads=8, concat) ----------------
    {
        const int tiles = (N_NODES / 16) * (H1DIM / 16);   // 25000 wave-tiles
        gat_gemm_wmma_f32<<<cdiv(tiles, 8), B, 0, stream>>>(x, W1, h1, N_NODES, H1DIM);
    }
    gat_attn_coef<HEADS><<<cdiv((long long)N_NODES * HEADS, B), B, 0, stream>>>(
        h1, as1, ad1, als1, ald1, emax1);
    gat_edge_max<HEADS><<<cdiv((long long)N_EDGES * HEADS, B), B, 0, stream>>>(
        src, dstp, als1, ald1, emax1);
    gat_init_sums<HEADS><<<cdiv((long long)N_NODES * HEADS, B), B, 0, stream>>>(
        h1, als1, ald1, emax1, denom1, num1);
    gat_edge_agg<HEADS><<<cdiv((long long)N_EDGES * HEADS, B), B, 0, stream>>>(
        src, dstp, h1, als1, ald1, emax1, denom1, num1);
    gat_finalize<HEADS><<<cdiv((long long)N_NODES * H1DIM, B), B, 0, stream>>>(
        num1, denom1, b1, num1);   // in-place -> relu'd layer-2 input

    // ---------------- layer 2: GATConv(128 -> 16, heads=1, concat=False) ----------------
    {
        const int tiles = (N_NODES / 16) * (NHID / 16);    // 3125 wave-tiles
        gat_gemm_wmma_f32<<<cdiv(tiles, 8), B, 0, stream>>>(num1, W2, h2, N_NODES, NHID);
    }
    gat_attn_coef<1><<<cdiv(N_NODES, B), B, 0, stream>>>(h2, as2, ad2, als2, ald2, emax2);
    gat_edge_max<1><<<cdiv(N_EDGES, B), B, 0, stream>>>(src, dstp, als2, ald2, emax2);
    gat_init_sums<1><<<cdiv(N_NODES, B), B, 0, stream>>>(h2, als2, ald2, emax2, denom2, num2);
    gat_edge_agg<1><<<cdiv(N_EDGES, B), B, 0, stream>>>(src, dstp, h2, als2, ald2, emax2,
                                                        denom2, num2);
    gat_finalize<1><<<cdiv((long long)N_NODES * NHID, B), B, 0, stream>>>(
        num2, denom2, b2, out);
}